// PixelContrastLoss_51075751084566
// MI455X (gfx1250) — compile-verified
//
#include <hip/hip_runtime.h>
#include <hip/hip_bf16.h>
#include <math.h>

// ---------------------------------------------------------------------------
// PixelContrastLoss on MI455X (gfx1250), wave32, fp32 WMMA (16x16x4).
//
// Pipeline (all deterministic, no global atomics):
//   1) select_topk    : exact radix-select top-512 per class (pix + queue)
//   2) gather         : Xa[4096][128] fp32 + diag = ||x||^2 / T
//   3) rowmax_part    : per-j-chunk row max (4 chunks, 1024 waves)
//   4) combine_rowmax
//   5) sums_part      : per-j-chunk Stot/Psame partial exp-sums
//   6) combine_sums
//   7) loss_part      : masked log-prob numerator over same-class columns
//   8) finalize       : scalar = mean loss
//
// GEMM tiles: A and B both staged in VGPR arrays (32 x float2 each) so the
// 32 global_load_b64 per tile issue as one pipelined clause; 4 independent
// WMMA accumulators break the D->C dependency chain.
// ---------------------------------------------------------------------------

#define NUM_CLASSES 4
#define MAX_VIEWS   512
#define QUEUE_SIZE  5000
#define DIM         128
#define N_PIX       (2 * 1 * 32 * 64 * 64)   // 262144
#define DHW         (32 * 64 * 64)           // 131072
#define N_ANCHOR    4096
#define INV_T       10.0f                    // 1/TEMPERATURE
#define T_OVER_BT   (0.1f / 0.07f)

typedef float  v2f __attribute__((ext_vector_type(2)));
typedef float  v8f __attribute__((ext_vector_type(8)));

// ---------------------------------------------------------------------------
// 1) Exact top-512 selection. Blocks 0..3: pixel classes, 4..7: queue classes.
// ---------------------------------------------------------------------------
__device__ __forceinline__ unsigned sel_key(int i, int is_pix, int cls,
                                            const int* labels,
                                            const float* pix_scores,
                                            const float* queue_scores) {
  if (is_pix) {
    unsigned k = __float_as_uint(pix_scores[i]);
    return (labels[i] == cls) ? k : 0u;
  } else {
    return __float_as_uint(queue_scores[cls * QUEUE_SIZE + i]);
  }
}

__global__ void select_topk_kernel(const int* __restrict__ labels,
                                   const float* __restrict__ pix_scores,
                                   const float* __restrict__ queue_scores,
                                   int* __restrict__ sel_pix,
                                   int* __restrict__ sel_q) {
  const int tid = threadIdx.x;                 // 256 threads
  const int blk = blockIdx.x;                  // 8 blocks
  const int is_pix = (blk < NUM_CLASSES) ? 1 : 0;
  const int cls = is_pix ? blk : (blk - NUM_CLASSES);
  const int n = is_pix ? N_PIX : QUEUE_SIZE;
  int* out = is_pix ? (sel_pix + cls * MAX_VIEWS) : (sel_q + cls * MAX_VIEWS);

  __shared__ unsigned hist[256];
  __shared__ unsigned scanbuf[256];
  __shared__ unsigned sh_pref, sh_pmask;
  __shared__ int sh_rem;
  __shared__ unsigned baseGt, baseEq;

  if (tid == 0) { sh_pref = 0u; sh_pmask = 0u; sh_rem = MAX_VIEWS; }
  __syncthreads();

  // ---- 4-pass radix select (MSB first) for the 512th-largest key ----
  for (int shift = 24; shift >= 0; shift -= 8) {
    hist[tid] = 0u;
    __syncthreads();
    const unsigned pref = sh_pref, pmask = sh_pmask;
    for (int i = tid; i < n; i += 256) {
      unsigned k = sel_key(i, is_pix, cls, labels, pix_scores, queue_scores);
      if ((k & pmask) == pref)
        atomicAdd(&hist[(k >> shift) & 255u], 1u);
    }
    __syncthreads();
    if (tid == 0) {
      unsigned cum = 0; int b = 255;
      for (; b > 0; --b) {
        unsigned h = hist[b];
        if (cum + h >= (unsigned)sh_rem) break;
        cum += h;
      }
      sh_pref |= ((unsigned)b) << shift;
      sh_pmask |= 255u << shift;
      sh_rem -= (int)cum;
    }
    __syncthreads();
  }

  const unsigned kth = sh_pref;
  const int remEq = sh_rem;                 // ties to take (lowest index first)
  const int countGt = MAX_VIEWS - remEq;    // number of strictly-greater keys

  if (tid == 0) { baseGt = 0u; baseEq = 0u; }
  __syncthreads();

  // ---- ordered compaction with 256-wide Hillis–Steele scan in LDS ----
  for (int start = 0; start < n; start += 256) {
    const int i = start + tid;
    unsigned k = 0u;
    if (i < n) k = sel_key(i, is_pix, cls, labels, pix_scores, queue_scores);
    const unsigned gt = (i < n && k > kth) ? 1u : 0u;
    const unsigned eq = (i < n && k == kth) ? 1u : 0u;
    const unsigned packed = (gt << 16) | eq;

    scanbuf[tid] = packed;
    __syncthreads();
    for (int off = 1; off < 256; off <<= 1) {
      unsigned add = (tid >= off) ? scanbuf[tid - off] : 0u;
      __syncthreads();
      scanbuf[tid] += add;
      __syncthreads();
    }
    const unsigned excl = scanbuf[tid] - packed;
    const unsigned tot = scanbuf[255];

    if (gt) out[baseGt + (excl >> 16)] = i;
    if (eq) {
      unsigned p = baseEq + (excl & 0xFFFFu);
      if (p < (unsigned)remEq) out[countGt + p] = i;
    }
    __syncthreads();
    if (tid == 0) { baseGt += tot >> 16; baseEq += tot & 0xFFFFu; }
    __syncthreads();
  }
}

// ---------------------------------------------------------------------------
// 2) Gather anchors + diag = ||x||^2 / T.
// ---------------------------------------------------------------------------
__global__ void gather_kernel(const float* __restrict__ feats,
                              const float* __restrict__ pixel_queue,
                              const int* __restrict__ sel_pix,
                              const int* __restrict__ sel_q,
                              float* __restrict__ Xa,
                              float* __restrict__ diag) {
  const int r = blockIdx.x;        // 4096
  const int ch = threadIdx.x;      // 128
  float v;
  if (r < 2048) {
    const int c = r >> 9, k = r & 511;
    const int nidx = sel_pix[c * MAX_VIEWS + k];
    const int outer = nidx / DHW;
    const int sp = nidx - outer * DHW;
    v = feats[(outer * DIM + ch) * DHW + sp];
  } else {
    const int rr = r - 2048;
    const int c = rr >> 9, k = rr & 511;
    const int q = sel_q[c * MAX_VIEWS + k];
    v = pixel_queue[(c * QUEUE_SIZE + q) * DIM + ch];
  }
  Xa[r * DIM + ch] = v;

  float s = v * v;
  #pragma unroll
  for (int m = 16; m >= 1; m >>= 1) s += __shfl_xor(s, m);
  __shared__ float wsum[4];
  if ((threadIdx.x & 31) == 0) wsum[threadIdx.x >> 5] = s;
  __syncthreads();
  if (threadIdx.x == 0)
    diag[r] = (wsum[0] + wsum[1] + wsum[2] + wsum[3]) * INV_T;
}

// ---------------------------------------------------------------------------
// Register-staged 16x16 tile of X @ X^T via V_WMMA_F32_16X16X4_F32.
// A layout (16x4 f32): lane L: M = L&15, VGPR{0,1} = K{0,1} (L<16) / K{2,3}.
// B layout (4x16 f32) mirrored: lane L: N = L&15, VGPR{0,1} = K{0,1}/{2,3}.
// Both reduce to a float2 load at Xa[(base + (L&15))*128 + 4*kk + 2*(L>>4)].
// ---------------------------------------------------------------------------
__device__ __forceinline__ void load_rowpairs(const float* __restrict__ Xa,
                                              int row0, int lane,
                                              v2f* __restrict__ dst) {
  const float* p = Xa + (row0 + (lane & 15)) * DIM + ((lane >> 4) << 1);
  #pragma unroll
  for (int kk = 0; kk < 32; ++kk) dst[kk] = *(const v2f*)(p + kk * 4);
}

__device__ __forceinline__ v8f gemm_tile_regs(const v2f* __restrict__ a,
                                              const v2f* __restrict__ b) {
  v8f c0 = {}, c1 = {}, c2 = {}, c3 = {};
  #pragma unroll
  for (int kk = 0; kk < 32; kk += 4) {
    c0 = __builtin_amdgcn_wmma_f32_16x16x4_f32(false, a[kk + 0], false,
                                               b[kk + 0], (short)0, c0,
                                               false, false);
    c1 = __builtin_amdgcn_wmma_f32_16x16x4_f32(false, a[kk + 1], false,
                                               b[kk + 1], (short)0, c1,
                                               false, false);
    c2 = __builtin_amdgcn_wmma_f32_16x16x4_f32(false, a[kk + 2], false,
                                               b[kk + 2], (short)0, c2,
                                               false, false);
    c3 = __builtin_amdgcn_wmma_f32_16x16x4_f32(false, a[kk + 3], false,
                                               b[kk + 3], (short)0, c3,
                                               false, false);
  }
  return (c0 + c1) + (c2 + c3);
}

// C/D layout: VGPR r, lanes 0-15 -> (M=r, N=lane), lanes 16-31 -> (M=r+8).
// Row reductions run across the 16-lane halves (xor masks 1,2,4,8).

// ---------------------------------------------------------------------------
// 3) Row max, split into 4 j-chunks of 64 tiles (1024 waves).
// ---------------------------------------------------------------------------
__global__ void rowmax_part_kernel(const float* __restrict__ Xa,
                                   float* __restrict__ part) {
  const int wid = blockIdx.x * (blockDim.x >> 5) + (threadIdx.x >> 5); // 0..1023
  const int ti = wid & 255;
  const int chunk = wid >> 8;
  const int lane = threadIdx.x & 31;
  const int hi = lane >> 4;

  v2f a[32];
  load_rowpairs(Xa, ti * 16, lane, a);

  float m[8];
  #pragma unroll
  for (int r = 0; r < 8; ++r) m[r] = -3.4e38f;

  for (int tj = chunk * 64; tj < chunk * 64 + 64; ++tj) {
    v2f b[32];
    load_rowpairs(Xa, tj * 16, lane, b);
    v8f acc = gemm_tile_regs(a, b);
    #pragma unroll
    for (int r = 0; r < 8; ++r) m[r] = fmaxf(m[r], acc[r] * INV_T);
  }
  #pragma unroll
  for (int r = 0; r < 8; ++r) {
    #pragma unroll
    for (int xm = 1; xm < 16; xm <<= 1)
      m[r] = fmaxf(m[r], __shfl_xor(m[r], xm));
  }
  if ((lane & 15) == 0) {
    #pragma unroll
    for (int r = 0; r < 8; ++r)
      part[chunk * N_ANCHOR + ti * 16 + r + 8 * hi] = m[r];
  }
}

__global__ void combine_rowmax_kernel(const float* __restrict__ part,
                                      float* __restrict__ rowmax) {
  const int i = blockIdx.x * blockDim.x + threadIdx.x;  // 4096
  float m = fmaxf(fmaxf(part[i], part[N_ANCHOR + i]),
                  fmaxf(part[2 * N_ANCHOR + i], part[3 * N_ANCHOR + i]));
  rowmax[i] = m;
}

// ---------------------------------------------------------------------------
// 5) Partial exp-sums (total and same-class incl. diag), 4 j-chunks.
// ---------------------------------------------------------------------------
__global__ void sums_part_kernel(const float* __restrict__ Xa,
                                 const float* __restrict__ rowmax,
                                 float* __restrict__ stot_part,
                                 float* __restrict__ psame_part) {
  const int wid = blockIdx.x * (blockDim.x >> 5) + (threadIdx.x >> 5);
  const int ti = wid & 255;
  const int chunk = wid >> 8;
  const int lane = threadIdx.x & 31;
  const int hi = lane >> 4;
  const int ci = ((ti * 16) & 2047) >> 9;

  v2f a[32];
  load_rowpairs(Xa, ti * 16, lane, a);

  float m[8], st[8], ps[8];
  #pragma unroll
  for (int r = 0; r < 8; ++r) {
    m[r] = rowmax[ti * 16 + r + 8 * hi];
    st[r] = 0.0f; ps[r] = 0.0f;
  }
  for (int tj = chunk * 64; tj < chunk * 64 + 64; ++tj) {
    v2f b[32];
    load_rowpairs(Xa, tj * 16, lane, b);
    v8f acc = gemm_tile_regs(a, b);
    const bool same = (((tj * 16) & 2047) >> 9) == ci;
    #pragma unroll
    for (int r = 0; r < 8; ++r) {
      float e = expf(acc[r] * INV_T - m[r]);
      st[r] += e;
      if (same) ps[r] += e;
    }
  }
  #pragma unroll
  for (int r = 0; r < 8; ++r) {
    #pragma unroll
    for (int xm = 1; xm < 16; xm <<= 1) {
      st[r] += __shfl_xor(st[r], xm);
      ps[r] += __shfl_xor(ps[r], xm);
    }
  }
  if ((lane & 15) == 0) {
    #pragma unroll
    for (int r = 0; r < 8; ++r) {
      const int row = ti * 16 + r + 8 * hi;
      stot_part[chunk * N_ANCHOR + row] = st[r];
      psame_part[chunk * N_ANCHOR + row] = ps[r];
    }
  }
}

__global__ void combine_sums_kernel(const float* __restrict__ stot_part,
                                    const float* __restrict__ psame_part,
                                    float* __restrict__ Stot,
                                    float* __restrict__ Psame) {
  const int i = blockIdx.x * blockDim.x + threadIdx.x;  // 4096
  Stot[i] = (stot_part[i] + stot_part[N_ANCHOR + i]) +
            (stot_part[2 * N_ANCHOR + i] + stot_part[3 * N_ANCHOR + i]);
  Psame[i] = (psame_part[i] + psame_part[N_ANCHOR + i]) +
             (psame_part[2 * N_ANCHOR + i] + psame_part[3 * N_ANCHOR + i]);
}

// ---------------------------------------------------------------------------
// 7) Numerator over same-class columns (2 x 512 cols = 64 tiles), split in
//    two 32-tile halves (512 waves).
//    neg_i = Stot_i - Psame_i + exp(l_ii - m_i)
//    num_i = sum_{same, j!=i} [ sh_ij - log(exp(sh_ij) + neg_i + 1e-10) ]
// ---------------------------------------------------------------------------
__global__ void loss_part_kernel(const float* __restrict__ Xa,
                                 const float* __restrict__ rowmax,
                                 const float* __restrict__ Stot,
                                 const float* __restrict__ Psame,
                                 const float* __restrict__ diag,
                                 float* __restrict__ num_part) {
  const int wid = blockIdx.x * (blockDim.x >> 5) + (threadIdx.x >> 5); // 0..511
  const int ti = wid & 255;
  const int half = wid >> 8;
  const int lane = threadIdx.x & 31;
  const int hi = lane >> 4;
  const int rlo = lane & 15;
  const int ci = ((ti * 16) & 2047) >> 9;

  v2f a[32];
  load_rowpairs(Xa, ti * 16, lane, a);

  float m[8], neg[8], num[8];
  #pragma unroll
  for (int r = 0; r < 8; ++r) {
    const int row = ti * 16 + r + 8 * hi;
    const float mm = rowmax[row];
    m[r] = mm;
    neg[r] = Stot[row] - Psame[row] + expf(diag[row] - mm);
    num[r] = 0.0f;
  }

  for (int s = half * 32; s < half * 32 + 32; ++s) {
    const int tj = (s < 32) ? (ci * 32 + s) : (128 + ci * 32 + (s - 32));
    v2f b[32];
    load_rowpairs(Xa, tj * 16, lane, b);
    v8f acc = gemm_tile_regs(a, b);
    #pragma unroll
    for (int r = 0; r < 8; ++r) {
      const float sh = acc[r] * INV_T - m[r];
      const float term = sh - logf(expf(sh) + neg[r] + 1e-10f);
      const bool isdiag = (tj == ti) && (rlo == r + 8 * hi);
      if (!isdiag) num[r] += term;
    }
  }
  #pragma unroll
  for (int r = 0; r < 8; ++r) {
    #pragma unroll
    for (int xm = 1; xm < 16; xm <<= 1) num[r] += __shfl_xor(num[r], xm);
  }
  if ((lane & 15) == 0) {
    #pragma unroll
    for (int r = 0; r < 8; ++r)
      num_part[half * N_ANCHOR + ti * 16 + r + 8 * hi] = num[r];
  }
}

// ---------------------------------------------------------------------------
// 8) Deterministic reduction: per-row loss -> mean over 4096 anchors.
// ---------------------------------------------------------------------------
__global__ void finalize_kernel(const float* __restrict__ num_part,
                                float* __restrict__ out) {
  __shared__ float s[256];
  const int tid = threadIdx.x;
  float acc = 0.0f;
  for (int i = tid; i < N_ANCHOR; i += 256) {
    const float num = num_part[i] + num_part[N_ANCHOR + i];
    acc += -T_OVER_BT * num / (1023.0f + 1e-10f);  // mask.sum(1) == 1023
  }
  s[tid] = acc;
  __syncthreads();
  for (int off = 128; off > 0; off >>= 1) {
    if (tid < off) s[tid] += s[tid + off];
    __syncthreads();
  }
  if (tid == 0) out[0] = s[0] / (float)N_ANCHOR;
}

// ---------------------------------------------------------------------------
extern "C" void kernel_launch(void* const* d_in, const int* in_sizes, int n_in,
                              void* d_out, int out_size, void* d_ws, size_t ws_size,
                              hipStream_t stream) {
  const float* feats        = (const float*)d_in[0];
  const int*   labels       = (const int*)  d_in[1];
  const float* pixel_queue  = (const float*)d_in[2];
  const float* pix_scores   = (const float*)d_in[3];
  const float* queue_scores = (const float*)d_in[4];
  float* out = (float*)d_out;

  // Workspace layout (~2.4 MB)
  float* Xa         = (float*)d_ws;                    // 4096*128
  float* diag       = Xa + N_ANCHOR * DIM;             // 4096
  float* rowmax     = diag + N_ANCHOR;                 // 4096
  float* Stot       = rowmax + N_ANCHOR;               // 4096
  float* Psame      = Stot + N_ANCHOR;                 // 4096
  float* rmax_part  = Psame + N_ANCHOR;                // 4*4096
  float* stot_part  = rmax_part + 4 * N_ANCHOR;        // 4*4096
  float* psame_part = stot_part + 4 * N_ANCHOR;        // 4*4096
  float* num_part   = psame_part + 4 * N_ANCHOR;       // 2*4096
  int*   sel_pix    = (int*)(num_part + 2 * N_ANCHOR); // 4*512
  int*   sel_q      = sel_pix + NUM_CLASSES * MAX_VIEWS;

  select_topk_kernel<<<8, 256, 0, stream>>>(labels, pix_scores, queue_scores,
                                            sel_pix, sel_q);
  gather_kernel<<<N_ANCHOR, DIM, 0, stream>>>(feats, pixel_queue, sel_pix,
                                              sel_q, Xa, diag);
  rowmax_part_kernel<<<256, 128, 0, stream>>>(Xa, rmax_part);
  combine_rowmax_kernel<<<16, 256, 0, stream>>>(rmax_part, rowmax);
  sums_part_kernel<<<256, 128, 0, stream>>>(Xa, rowmax, stot_part, psame_part);
  combine_sums_kernel<<<16, 256, 0, stream>>>(stot_part, psame_part, Stot,
                                              Psame);
  loss_part_kernel<<<128, 128, 0, stream>>>(Xa, rowmax, Stot, Psame, diag,
                                            num_part);
  finalize_kernel<<<1, 256, 0, stream>>>(num_part, out);
}